// GCNN_26688926777484
// MI455X (gfx1250) — compile-verified
//
#include <hip/hip_runtime.h>
#include <hip/hip_bf16.h>

// GCN two-branch network for MI455X (gfx1250, wave32).
// Dominant compute: two 10000x1024x1024 fp32 GEMMs -> V_WMMA_F32_16X16X4_F32
// (exact fp32 matrix pipe; bf16 WMMA would degrade fp32-reference accuracy).
// GEMM: per-block LDS-staged B tiles (pair-interleaved -> ds_load_b64
// fragments, bank-padded), 16x64 output strip per wave (4 accumulators) so
// each A fragment feeds 4 WMMAs and A L2 traffic drops 4x.
// Dominant memory: 80000-edge gather/scatter-add (~650 MB/branch) -> f32
// global atomics, HBM-bound (~30us @ 23.3 TB/s).

#define F_DIM 1024
#define NUM_GRAPHS 64
#define NEG_SLOPE 0.01f
#define WAVES_PER_BLOCK 8
#define KC 32            // GEMM K-chunk staged in LDS
#define PROW 80          // LDS pair-row stride in float2 (160 f32 = 32 mod 64 banks)

typedef __attribute__((ext_vector_type(2))) float v2f;
typedef __attribute__((ext_vector_type(8))) float v8f;

__device__ __forceinline__ float leaky(float v) {
    return v >= 0.0f ? v : NEG_SLOPE * v;
}

// ---------------------------------------------------------------- utilities
__global__ void fill_f32(float* __restrict__ p, float v, int n) {
    int i = blockIdx.x * blockDim.x + threadIdx.x;
    if (i < n) p[i] = v;
}

__global__ void edge_degree(const int* __restrict__ dst, float* __restrict__ deg, int E) {
    int e = blockIdx.x * blockDim.x + threadIdx.x;
    if (e < E) atomicAdd(&deg[dst[e]], 1.0f);
}

__global__ void compute_dinv(float* __restrict__ deg, int n) {
    int i = blockIdx.x * blockDim.x + threadIdx.x;
    if (i < n) {
        float d = deg[i];
        deg[i] = d > 0.0f ? rsqrtf(d) : 0.0f;  // becomes dinv in-place
    }
}

// ----------------------------------------------------- fp32 WMMA GEMM C=A*B
// A: [M,K] row-major, B: [K,N] row-major, C: [M,N]. N multiple of 64, M
// multiple of 16 (M>=16), K multiple of 32. Block = 8 waves covering a
// 128-row x 64-col tile; each wave owns a 16x64 strip (4 WMMA accumulators).
// B K-chunk (32x64) is staged in LDS pair-interleaved so every fragment is
// one ds_load_b64; pair-row stride padded so lane-halves use disjoint banks.
// Fragment layouts per CDNA5 ISA 7.12.2:
//   A 16x4 f32 : lanes 0-15 M=0..15 K={0,1}; lanes 16-31 M=0..15 K={2,3}
//   B 4x16 f32 : lanes 0-15 N=0..15 K={0,1}; lanes 16-31 N=0..15 K={2,3}
//   C/D 16x16  : VGPR r -> M=r (lanes 0-15) / M=8+r (lanes 16-31), N=lane%16
__global__ void __launch_bounds__(256)
gemm_wmma_f32(const float* __restrict__ A, const float* __restrict__ B,
              const float* __restrict__ bias, float* __restrict__ C,
              int M, int N, int K, int act) {
    __shared__ float ldsB[(KC / 2) * PROW * 2];   // 16 pair-rows x 80 float2

    const int tid  = threadIdx.x;
    const int lane = tid & 31;
    const int wave = tid >> 5;
    const int half = lane >> 4;   // 0: K pair {0,1}; 1: K pair {2,3}
    const int l    = lane & 15;

    const int nb = blockIdx.x * 64;                        // block column base
    int m0 = (blockIdx.y * WAVES_PER_BLOCK + wave) * 16;   // wave row base
    if (m0 > M - 16) m0 = M - 16;  // clamp: duplicate identical work/stores

    const float* aptr = A + (size_t)(m0 + l) * K + 2 * half;  // 8B aligned

    v8f acc[4] = {};
    for (int k0 = 0; k0 < K; k0 += KC) {
        __syncthreads();
        // cooperative stage: B[k0..k0+31][nb..nb+63] -> pair-interleaved LDS
#pragma unroll
        for (int j = 0; j < (KC * 64) / 256; ++j) {
            int e = tid + 256 * j;
            int r = e >> 6, c = e & 63;                    // 256B-coalesced rows
            ldsB[(r >> 1) * (2 * PROW) + 2 * c + (r & 1)] =
                B[(size_t)(k0 + r) * N + nb + c];
        }
        __syncthreads();
#pragma unroll
        for (int s = 0; s < KC / 4; ++s) {
            float2 av = *(const float2*)(aptr + k0 + 4 * s);
            v2f a; a.x = av.x; a.y = av.y;
            const float2* brow = (const float2*)ldsB + (2 * s + half) * PROW;
#pragma unroll
            for (int nt = 0; nt < 4; ++nt) {
                float2 bv = brow[nt * 16 + l];             // ds_load_b64
                v2f b; b.x = bv.x; b.y = bv.y;
                // (neg_a, A, neg_b, B, c_mod, C, reuse_a, reuse_b)
                acc[nt] = __builtin_amdgcn_wmma_f32_16x16x4_f32(
                    false, a, false, b, (short)0, acc[nt], false, false);
            }
        }
    }

    const int mrow = m0 + half * 8;
#pragma unroll
    for (int nt = 0; nt < 4; ++nt) {
        const int col = nb + nt * 16 + l;
        const float bv = (bias != nullptr) ? bias[col] : 0.0f;
#pragma unroll
        for (int r = 0; r < 8; ++r) {
            float v = acc[nt][r] + bv;
            if (act) v = leaky(v);
            C[(size_t)(mrow + r) * N + col] = v;
        }
    }
}

// ------------------------------------------------------- edge scatter-add
// One block per edge; 256 threads stride the 1024 features.
__global__ void edge_aggregate(const float* __restrict__ h, const int* __restrict__ src,
                               const int* __restrict__ dst, const float* __restrict__ dinv,
                               float* __restrict__ agg, int E) {
    int e = blockIdx.x;
    if (e >= E) return;
    int s = src[e], d = dst[e];
    float w = dinv[s] * dinv[d];
    const float* hrow = h + (size_t)s * F_DIM;
    float* orow = agg + (size_t)d * F_DIM;
    for (int f = threadIdx.x; f < F_DIM; f += blockDim.x)
        atomicAdd(&orow[f], hrow[f] * w);
}

// self-loop term + bias + LeakyReLU, then scatter into per-graph pool sums.
__global__ void finalize_pool(const float* __restrict__ agg, const float* __restrict__ h,
                              const float* __restrict__ dinv, const float* __restrict__ bias,
                              const int* __restrict__ batch, float* __restrict__ pooled,
                              int Nn) {
    int i = blockIdx.x;
    if (i >= Nn) return;
    float di = dinv[i];
    float selfw = di * di;
    int g = batch[i];
    const float* arow = agg + (size_t)i * F_DIM;
    const float* hrow = h + (size_t)i * F_DIM;
    float* prow = pooled + (size_t)g * F_DIM;
    for (int f = threadIdx.x; f < F_DIM; f += blockDim.x) {
        float v = leaky(arow[f] + hrow[f] * selfw + bias[f]);
        atomicAdd(&prow[f], v);
    }
}

__global__ void count_nodes(const int* __restrict__ batch, float* __restrict__ cnt, int Nn) {
    int i = blockIdx.x * blockDim.x + threadIdx.x;
    if (i < Nn) atomicAdd(&cnt[batch[i]], 1.0f);
}

__global__ void mean_div(float* __restrict__ pooled, const float* __restrict__ cnt,
                         int B, int Fd) {
    int i = blockIdx.x * blockDim.x + threadIdx.x;
    if (i < B * Fd) pooled[i] /= fmaxf(cnt[i / Fd], 1.0f);
}

// out[g] = final_b + <concat(fc1[g],fc2[g]), final_W>  (256 MACs/graph)
__global__ void final_head(const float* __restrict__ fc1, const float* __restrict__ fc2,
                           const float* __restrict__ fw, const float* __restrict__ fb,
                           float* __restrict__ out, int B, int D) {
    int g = blockIdx.x * blockDim.x + threadIdx.x;
    if (g >= B) return;
    float s = fb[0];
    for (int d = 0; d < D; ++d) s += fc1[(size_t)g * D + d] * fw[d];
    for (int d = 0; d < D; ++d) s += fc2[(size_t)g * D + d] * fw[D + d];
    out[g] = s;
}

// ---------------------------------------------------------------- launcher
extern "C" void kernel_launch(void* const* d_in, const int* in_sizes, int n_in,
                              void* d_out, int out_size, void* d_ws, size_t ws_size,
                              hipStream_t stream) {
    (void)n_in; (void)out_size; (void)ws_size;
    const int F  = F_DIM;
    const int Nn = in_sizes[0] / F;   // 10000
    const int E  = in_sizes[1] / 2;   // 80000
    const int B  = NUM_GRAPHS;        // 64
    const int D  = in_sizes[13];      // 128

    const float* x[2]     = {(const float*)d_in[0],  (const float*)d_in[3]};
    const int*   ei[2]    = {(const int*)d_in[1],    (const int*)d_in[4]};
    const int*   batch[2] = {(const int*)d_in[2],    (const int*)d_in[5]};
    const float* convW[2] = {(const float*)d_in[10], (const float*)d_in[14]};
    const float* convB[2] = {(const float*)d_in[11], (const float*)d_in[15]};
    const float* fcW[2]   = {(const float*)d_in[12], (const float*)d_in[16]};
    const float* fcB[2]   = {(const float*)d_in[13], (const float*)d_in[17]};
    const float* finW     = (const float*)d_in[18];
    const float* finB     = (const float*)d_in[19];

    // scratch layout (floats), reused across the two branches (~83 MB total)
    float* ws = (float*)d_ws;
    size_t off = 0;
    auto alloc = [&](size_t n) { float* p = ws + off; off += (n + 255) & ~(size_t)255; return p; };
    float* h      = alloc((size_t)Nn * F);
    float* agg    = alloc((size_t)Nn * F);
    float* deg    = alloc((size_t)Nn);      // becomes dinv in-place
    float* pooled = alloc((size_t)B * F);
    float* cnt    = alloc((size_t)B);
    float* fcout0 = alloc((size_t)B * D);
    float* fcout1 = alloc((size_t)B * D);
    float* fcout[2] = {fcout0, fcout1};

    const dim3 blk256(256);
    for (int br = 0; br < 2; ++br) {
        // symmetric-norm degrees (init 1.0 == self-loop), then deg^-1/2
        fill_f32<<<(Nn + 255) / 256, blk256, 0, stream>>>(deg, 1.0f, Nn);
        edge_degree<<<(E + 255) / 256, blk256, 0, stream>>>(ei[br] + E, deg, E);
        compute_dinv<<<(Nn + 255) / 256, blk256, 0, stream>>>(deg, Nn);

        // h = x @ W  (fp32 WMMA; block tile 128x64, wave strip 16x64)
        {
            dim3 grid(F / 64, (Nn / 16 + WAVES_PER_BLOCK - 1) / WAVES_PER_BLOCK);
            gemm_wmma_f32<<<grid, dim3(32 * WAVES_PER_BLOCK), 0, stream>>>(
                x[br], convW[br], nullptr, h, Nn, F, F, 0);
        }

        // message passing: agg[dst] += h[src] * dinv[src]*dinv[dst]
        fill_f32<<<(Nn * F + 255) / 256, blk256, 0, stream>>>(agg, 0.0f, Nn * F);
        edge_aggregate<<<E, blk256, 0, stream>>>(h, ei[br], ei[br] + E, deg, agg, E);

        // activation + global mean pool
        fill_f32<<<(B * F + 255) / 256, blk256, 0, stream>>>(pooled, 0.0f, B * F);
        fill_f32<<<1, blk256, 0, stream>>>(cnt, 0.0f, B);
        count_nodes<<<(Nn + 255) / 256, blk256, 0, stream>>>(batch[br], cnt, Nn);
        finalize_pool<<<Nn, blk256, 0, stream>>>(agg, h, deg, convB[br], batch[br], pooled, Nn);
        mean_div<<<(B * F + 255) / 256, blk256, 0, stream>>>(pooled, cnt, B, F);

        // fc: leaky([B,F] @ [F,D] + b)  (fp32 WMMA, fused epilogue)
        {
            dim3 grid(D / 64, 1);
            gemm_wmma_f32<<<grid, dim3(32 * WAVES_PER_BLOCK), 0, stream>>>(
                pooled, fcW[br], fcB[br], fcout[br], B, D, F, 1);
        }
    }

    final_head<<<1, 64, 0, stream>>>(fcout[0], fcout[1], finW, finB, (float*)d_out, B, D);
}